// Photodetector_45140106281489
// MI455X (gfx1250) — compile-verified
//
#include <hip/hip_runtime.h>
#include <math.h>

// ---------------- constants from the reference ----------------
#define QC      1.602176563e-19
#define KBC     1.3806488e-23
#define TKELV   300.0
#define BITRATE 5.0e10
#define DARKI   1.0e-10
#define LOADR   1.0e6
#define SHOTC   (2.0 * QC * BITRATE)                      // 2*Q*B
#define THERMC  (4.0 * KBC * TKELV * BITRATE / LOADR)     // 4kTB/R

#define TSTEPS 16384
#define BCOLS  512
#define NT     16388   // T + K - 1, K = 5

// workspace layout (in floats)
#define NSD_OFF 0        // [0, 16388)   noise_sd as f32
#define BC_OFF  16392    // 5 floats     FIR coefs f32
#define H_OFF   16400    // 16 floats    AR impulse response
#define G_OFF   16416    // 64 floats    state-propagation g[t][p]

typedef __attribute__((ext_vector_type(2))) float v2f;
typedef __attribute__((ext_vector_type(8))) float v8f;

// ---------------------------------------------------------------
// Kernel 1: per-time-row batch mean of signal -> noise_sd (f32)
// ---------------------------------------------------------------
__global__ void __launch_bounds__(256) rowsum_nsd(const float* __restrict__ sig,
                                                  float* __restrict__ nsd) {
    __shared__ double red[256];
    const int r   = blockIdx.x;            // time row 0..16383
    const int tid = threadIdx.x;
    const float* row = sig + (size_t)r * BCOLS;
    red[tid] = (double)row[tid] + (double)row[tid + 256];
    __syncthreads();
    #pragma unroll
    for (int off = 128; off > 0; off >>= 1) {
        if (tid < off) red[tid] += red[tid + off];
        __syncthreads();
    }
    if (tid == 0) {
        double mean = red[0] * (1.0 / (double)BCOLS);
        nsd[r + 4] = (float)sqrt(SHOTC * (mean + DARKI) + THERMC);
    }
    if (r == 0 && tid < 4) {  // padded region: mean_sig == 0
        nsd[tid] = (float)sqrt(SHOTC * DARKI + THERMC);
    }
}

// ---------------------------------------------------------------
// Kernel 2: coefficient tables (f64 -> f32):
//   bc[k]            FIR taps
//   h[0..15]         AR impulse response: h[0]=1, h[t]=-sum a[j] h[t-j]
//   g[t][p]          response at step t to initial state y[-4+p]=1
// ---------------------------------------------------------------
__global__ void setup_coefs(const double* __restrict__ b,
                            const double* __restrict__ a_rev,
                            float* __restrict__ wsf) {
    if (threadIdx.x != 0 || blockIdx.x != 0) return;
    double a[5];
    for (int i = 0; i < 5; ++i) a[i] = a_rev[4 - i];   // a[0]=1, a[1..4]
    for (int k = 0; k < 5; ++k) wsf[BC_OFF + k] = (float)b[k];

    double h[16];
    h[0] = 1.0;
    for (int t = 1; t < 16; ++t) {
        double acc = 0.0;
        int jm = t < 4 ? t : 4;
        for (int j = 1; j <= jm; ++j) acc -= a[j] * h[t - j];
        h[t] = acc;
    }
    for (int t = 0; t < 16; ++t) wsf[H_OFF + t] = (float)h[t];

    double g[16][4];
    for (int t = 0; t < 16; ++t) {
        for (int p = 0; p < 4; ++p) {
            int jd = t - p + 4;                       // delta hit: j == t-p+4
            double acc = (jd >= 1 && jd <= 4) ? -a[jd] : 0.0;
            int jm = t < 4 ? t : 4;
            for (int j = 1; j <= jm; ++j) acc -= a[j] * g[t - j][p];
            g[t][p] = acc;
        }
    }
    for (int t = 0; t < 16; ++t)
        for (int p = 0; p < 4; ++p)
            wsf[G_OFF + t * 4 + p] = (float)g[t][p];
}

// ---------------------------------------------------------------
// Kernel 3: fused noise + FIR + blocked IIR via WMMA.
// One wave per 16-column tile (32 waves total). Per 16-step block:
//   y_block(16x16) = H(16x16)*s_block(16x16) + G(16x4)*state(4x16)
// as 5x V_WMMA_F32_16X16X4_F32 accumulating into one v8f.
// ---------------------------------------------------------------
__global__ void __launch_bounds__(256) fir_iir_wmma(const float* __restrict__ sig,
                                                    const float* __restrict__ noise,
                                                    const float* __restrict__ wsf,
                                                    float* __restrict__ out) {
    const int lane = threadIdx.x & 31;
    const int wid  = blockIdx.x * (blockDim.x >> 5) + (threadIdx.x >> 5);
    const int c0   = wid * 16;          // column tile base
    const int n    = lane & 15;         // N index (column within tile)
    const int hi   = lane >> 4;         // half-wave select
    const int d    = hi * 2;            // K sub-row offset for A/B layout
    const int cg   = c0 + n;            // global column

    const float* nsd = wsf + NSD_OFF;
    const float* hp  = wsf + H_OFF;
    const float* gp  = wsf + G_OFF;

    float bc0 = wsf[BC_OFF + 0], bc1 = wsf[BC_OFF + 1], bc2 = wsf[BC_OFF + 2],
          bc3 = wsf[BC_OFF + 3], bc4 = wsf[BC_OFF + 4];

    // A operands (16x4 f32 layout: lane m=lane&15; VGPR0:K=k0, VGPR1:K=k0+1,
    // k0 = 2*(lane>>4) (+4j per K-slice)).
    const int m = n;
    v2f Ah[4];
    #pragma unroll
    for (int j = 0; j < 4; ++j) {
        int k0 = 4 * j + d;
        Ah[j].x = (m >= k0)     ? hp[m - k0]     : 0.0f;
        Ah[j].y = (m >= k0 + 1) ? hp[m - k0 - 1] : 0.0f;
    }
    v2f Ag;
    Ag.x = gp[m * 4 + d];
    Ag.y = gp[m * 4 + d + 1];

    // Recover a[1..4] for the scalar block-0 path: a[k] = -g[0][4-k]
    const float af1 = -gp[3], af2 = -gp[2], af3 = -gp[1], af4 = -gp[0];

    // ---------------- block 0 (scalar; pass-through rows 0..4) ----------
    float y12v, y13v, y14v, y15v;
    {
        float xn[20];
        #pragma unroll
        for (int i = 0; i < 20; ++i) {
            float xv = nsd[i] * noise[(size_t)cg * NT + i];
            if (i >= 4) xv += sig[(size_t)(i - 4) * BCOLS + cg];
            xn[i] = xv;
        }
        float y[16];
        #pragma unroll
        for (int t = 0; t < 16; ++t) {
            float sv = bc0 * xn[t] + bc1 * xn[t + 1] + bc2 * xn[t + 2] +
                       bc3 * xn[t + 3] + bc4 * xn[t + 4];
            y[t] = (t < 5) ? sv
                           : sv - (af1 * y[t - 1] + af2 * y[t - 2] +
                                   af3 * y[t - 3] + af4 * y[t - 4]);
        }
        if (lane < 16) {
            #pragma unroll
            for (int t = 0; t < 16; ++t) out[(size_t)t * BCOLS + cg] = y[t];
        }
        // lanes L and L+16 share cg, so state values are valid on all lanes
        y12v = y[12]; y13v = y[13]; y14v = y[14]; y15v = y[15];
    }

    // state in B-matrix (4x16) layout: VGPRv: lanes0-15 -> K=v, lanes16-31 -> K=v+2
    v2f Bst;
    Bst.x = hi ? y14v : y12v;
    Bst.y = hi ? y15v : y13v;

    // ---------------- blocks 1..1023 (WMMA) -----------------------------
    for (int blk = 1; blk < TSTEPS / 16; ++blk) {
        const int t0 = blk * 16;
        const float4* nz4 = (const float4*)(noise + (size_t)cg * NT + t0);
        const float4* sd4 = (const float4*)(nsd + t0);

        // prefetch next block's noise row (global_prefetch_b8)
        __builtin_prefetch(noise + (size_t)cg * NT + t0 + 16, 0, 1);

        float xn[20];
        #pragma unroll
        for (int i = 0; i < 5; ++i) {
            float4 nz = nz4[i];
            float4 sd = sd4[i];
            const size_t mrow = (size_t)(t0 - 4 + 4 * i);
            xn[4 * i + 0] = sd.x * nz.x + sig[(mrow + 0) * BCOLS + cg];
            xn[4 * i + 1] = sd.y * nz.y + sig[(mrow + 1) * BCOLS + cg];
            xn[4 * i + 2] = sd.z * nz.z + sig[(mrow + 2) * BCOLS + cg];
            xn[4 * i + 3] = sd.w * nz.w + sig[(mrow + 3) * BCOLS + cg];
        }

        // FIR: all 16 rows, static indices (stays in VGPRs)
        float s[16];
        #pragma unroll
        for (int r = 0; r < 16; ++r)
            s[r] = bc0 * xn[r] + bc1 * xn[r + 1] + bc2 * xn[r + 2] +
                   bc3 * xn[r + 3] + bc4 * xn[r + 4];

        // B operands for s_block: per half-wave pick rows {4j+d, 4j+d+1}
        v2f Bb[4];
        #pragma unroll
        for (int j = 0; j < 4; ++j) {
            Bb[j].x = hi ? s[4 * j + 2] : s[4 * j + 0];
            Bb[j].y = hi ? s[4 * j + 3] : s[4 * j + 1];
        }

        // y_block = G*state + sum_j H[:,4j:4j+4]*s[4j:4j+4,:]
        v8f acc = {0.f, 0.f, 0.f, 0.f, 0.f, 0.f, 0.f, 0.f};
        acc = __builtin_amdgcn_wmma_f32_16x16x4_f32(false, Ag, false, Bst,
                                                    (short)0, acc, false, false);
        #pragma unroll
        for (int j = 0; j < 4; ++j)
            acc = __builtin_amdgcn_wmma_f32_16x16x4_f32(false, Ah[j], false, Bb[j],
                                                        (short)0, acc, false, false);

        // store: D layout -> VGPRv: lanes0-15 row t0+v, lanes16-31 row t0+v+8
        #pragma unroll
        for (int v = 0; v < 8; ++v)
            out[(size_t)(t0 + v + 8 * hi) * BCOLS + cg] = acc[v];

        // next state = y rows 12..15 -> B layout via one cross-half shuffle
        const int src = 16 | n;                 // lane holding (row M=12+*, col n)
        float s4v = __shfl(acc[4], src, 32);    // y[12] col n
        float s5v = __shfl(acc[5], src, 32);    // y[13] col n
        Bst.x = hi ? acc[6] : s4v;              // hi lanes: y[14] already local
        Bst.y = hi ? acc[7] : s5v;              // hi lanes: y[15] already local
    }
}

// ---------------------------------------------------------------
extern "C" void kernel_launch(void* const* d_in, const int* in_sizes, int n_in,
                              void* d_out, int out_size, void* d_ws, size_t ws_size,
                              hipStream_t stream) {
    const float*  sig   = (const float*)d_in[0];   // (16384, 512) f32
    const float*  noise = (const float*)d_in[1];   // (512, 1, 16388) f32
    const double* bco   = (const double*)d_in[2];  // (5,) f64
    const double* arev  = (const double*)d_in[3];  // (5,) f64
    float* out = (float*)d_out;                    // (16384, 512) f32
    float* wsf = (float*)d_ws;

    rowsum_nsd<<<TSTEPS, 256, 0, stream>>>(sig, wsf);
    setup_coefs<<<1, 32, 0, stream>>>(bco, arev, wsf);
    fir_iir_wmma<<<4, 256, 0, stream>>>(sig, noise, wsf, out);
}